// HeteroGNN_5145370820876
// MI455X (gfx1250) — compile-verified
//
#include <hip/hip_runtime.h>
#include <hip/hip_bf16.h>
#include <stdint.h>

#define N_NODES 200000
#define E_EDGES 800000
#define DIM 64
#define N_REL 14
#define N_LAYERS 3

// pair-interleaved LDS layout for B operands:
//   element (k, c) of a 64x64 row-major weight lives at
//   mat*MAT_STRIDE + (k>>1)*PAIR_STRIDE + c*2 + (k&1)
// so the WMMA B fragment {W[k][c], W[k+1][c]} is one aligned b64 load.
// PAIR_STRIDE=160 (not 128) => upper half-wave (+1 pair row) lands on banks
// shifted by 32, disjoint from the lower half-wave for every n-tile.
#define PAIR_STRIDE 160
#define MAT_STRIDE  (32 * PAIR_STRIDE)   // 5120 floats per matrix

typedef __attribute__((ext_vector_type(2))) float v2f;
typedef __attribute__((ext_vector_type(4))) float v4f;
typedef __attribute__((ext_vector_type(8))) float v8f;

// ---------------------------------------------------------------------------
// Scatter: agg[dst, c] += x_src[src, c].  64 consecutive threads per edge so
// each wave issues coalesced 128B gathers and native global_atomic_add_f32.
// ---------------------------------------------------------------------------
__global__ __launch_bounds__(256) void scatter_sum_kernel(
    const float* __restrict__ xsrc,
    const long long* __restrict__ eidx,   // [2][E] int64 (src row, dst row)
    float* __restrict__ agg, int E)
{
  long long gtid = (long long)blockIdx.x * blockDim.x + threadIdx.x;
  long long e = gtid >> 6;
  int c = (int)(gtid & 63);
  if (e >= E) return;
  long long s = eidx[e];
  long long d = eidx[(long long)E + e];
  float v = xsrc[s * DIM + c];
  unsafeAtomicAdd(&agg[d * DIM + c], v);   // global_atomic_add_f32, no return
}

// ---------------------------------------------------------------------------
// Dense update: x_d = relu(agg @ Wr + b + x_d @ Wt), all fp32 via
// V_WMMA_F32_16X16X4_F32.  One wave per 16-row stripe; 4 x v8f accumulators
// cover all 64 output columns; weights pair-interleaved in LDS so every B
// fragment is a single conflict-free ds_load_b64.
//
// A frag (16x4, MxK): lane<16 -> {A[M=l15][k0], A[M=l15][k0+1]} with k0=4kc,
//                     lane>=16 -> same rows, k0=4kc+2.
// B frag (4x16, KxN): lane<16 -> {B[k0][col], B[k0+1][col]}, upper half k0+2.
// C/D (16x16): vgpr j, lane<16 -> (M=j, N=l15); lane>=16 -> (M=j+8, N=l15).
// ---------------------------------------------------------------------------
__global__ __launch_bounds__(256) void graphconv_update_kernel(
    const float* __restrict__ agg,
    float* __restrict__ x,                 // dst features [N,64], updated in place
    const float* __restrict__ Wr,          // [64,64] row-major (k, n)
    const float* __restrict__ Wt,          // [64,64] row-major (k, n)
    const float* __restrict__ bias,        // [64]
    int n_nodes)
{
  __shared__ float wlds[2 * MAT_STRIDE];
  {
    // 2048 (pair, col) slots per matrix; consecutive threads -> consecutive
    // cols, so both global 4B loads per slot are coalesced across the wave.
    for (int i = threadIdx.x; i < 2048; i += 256) {
      int p = i >> 6;          // k pair: k = 2p, 2p+1
      int c = i & 63;          // column
      v2f wr, wt;
      wr.x = Wr[(2 * p) * DIM + c];
      wr.y = Wr[(2 * p + 1) * DIM + c];
      wt.x = Wt[(2 * p) * DIM + c];
      wt.y = Wt[(2 * p + 1) * DIM + c];
      *(v2f*)(wlds + p * PAIR_STRIDE + c * 2)              = wr;
      *(v2f*)(wlds + MAT_STRIDE + p * PAIR_STRIDE + c * 2) = wt;
    }
  }
  __syncthreads();

  const int wave = threadIdx.x >> 5;
  const int lane = threadIdx.x & 31;
  const int half = lane >> 4;        // 0: lanes 0-15, 1: lanes 16-31
  const int l15  = lane & 15;
  const int m0 = blockIdx.x * 128 + wave * 16;
  if (m0 >= n_nodes) return;         // whole-wave uniform: EXEC all-1s inside

  v8f cacc[4];
  #pragma unroll
  for (int nt = 0; nt < 4; ++nt) {
    float b = bias[nt * 16 + l15];
    #pragma unroll
    for (int j = 0; j < 8; ++j) cacc[nt][j] = b;
  }

  const size_t rowbase = (size_t)(m0 + l15) * DIM;
  #pragma unroll 4
  for (int kc = 0; kc < 16; ++kc) {
    const int kbase = kc * 4 + half * 2;       // K start for this half-wave
    const int p     = 2 * kc + half;           // pair row in LDS
    v2f aR = *(const v2f*)(agg + rowbase + kbase);
    v2f aX = *(const v2f*)(x   + rowbase + kbase);
    const float* ldsR = wlds + p * PAIR_STRIDE;
    const float* ldsX = ldsR + MAT_STRIDE;
    #pragma unroll
    for (int nt = 0; nt < 4; ++nt) {
      const int co2 = (nt * 16 + l15) * 2;
      v2f bR = *(const v2f*)(ldsR + co2);
      v2f bX = *(const v2f*)(ldsX + co2);
      cacc[nt] = __builtin_amdgcn_wmma_f32_16x16x4_f32(
          false, aR, false, bR, (short)0, cacc[nt], false, false);
      cacc[nt] = __builtin_amdgcn_wmma_f32_16x16x4_f32(
          false, aX, false, bX, (short)0, cacc[nt], false, false);
    }
  }

  // fused ReLU + in-place store (this wave owns rows m0..m0+15 exclusively)
  #pragma unroll
  for (int nt = 0; nt < 4; ++nt) {
    const int col = nt * 16 + l15;
    #pragma unroll
    for (int j = 0; j < 8; ++j) {
      const int m = m0 + j + half * 8;
      float v = cacc[nt][j];
      x[(size_t)m * DIM + col] = v > 0.0f ? v : 0.0f;
    }
  }
}

// ---------------------------------------------------------------------------
// Host: d_out is the working feature store (6 slices); d_ws is agg scratch.
// 42 relation steps run back-to-back on `stream` to honor the sequential
// overwrite semantics of the reference.
// ---------------------------------------------------------------------------
extern "C" void kernel_launch(void* const* d_in, const int* in_sizes, int n_in,
                              void* d_out, int out_size, void* d_ws, size_t ws_size,
                              hipStream_t stream)
{
  static const int conns[N_REL][2] = {
      {0,0},{0,1},{1,0},{2,1},{3,2},{2,2},{2,4},
      {2,5},{3,5},{5,0},{5,2},{5,4},{4,1},{3,1}};

  const float*     W_rel      = (const float*)d_in[6];
  const float*     b_rel      = (const float*)d_in[7];
  const float*     W_root     = (const float*)d_in[8];
  const long long* edge_index = (const long long*)d_in[9];

  float* xs[6];
  for (int t = 0; t < 6; ++t) {
    xs[t] = (float*)d_out + (size_t)t * N_NODES * DIM;
    hipMemcpyAsync(xs[t], d_in[t], (size_t)N_NODES * DIM * sizeof(float),
                   hipMemcpyDeviceToDevice, stream);
  }
  float* agg = (float*)d_ws;   // needs N*D*4 = 51.2 MB of workspace

  const int scat_blocks = (int)(((long long)E_EDGES * DIM + 255) / 256);
  const int upd_blocks  = (N_NODES + 127) / 128;

  for (int layer = 0; layer < N_LAYERS; ++layer) {
    for (int r = 0; r < N_REL; ++r) {
      const int s = conns[r][0];
      const int d = conns[r][1];
      hipMemsetAsync(agg, 0, (size_t)N_NODES * DIM * sizeof(float), stream);
      scatter_sum_kernel<<<scat_blocks, 256, 0, stream>>>(
          xs[s], edge_index + (size_t)r * 2 * E_EDGES, agg, E_EDGES);
      graphconv_update_kernel<<<upd_blocks, 256, 0, stream>>>(
          agg, xs[d],
          W_rel  + (size_t)r * DIM * DIM,
          W_root + (size_t)r * DIM * DIM,
          b_rel  + (size_t)r * DIM,
          N_NODES);
    }
  }
}